// linearClassifier_bEF_83468394430957
// MI455X (gfx1250) — compile-verified
//
#include <hip/hip_runtime.h>

typedef __attribute__((ext_vector_type(2))) float v2f;
typedef __attribute__((ext_vector_type(4))) float v4f;
typedef __attribute__((ext_vector_type(8))) float v8f;

#define ND            8
#define EMB           128
#define DTOT          1024                 // ND * EMB, dot length per row
#define ROWS_PER_BLK  128                  // 8 waves x 16 rows
#define KTILE         64                   // K-tile staged through LDS
#define LDS_STRIDE    68                   // KTILE + 4 pad -> conflict-free WMMA reads
#define NKTILES       (DTOT / KTILE)       // 16
#define CHUNKS        (KTILE / 4)          // 16 WMMA K=4 steps per tile

// ---------------------------------------------------------------------------
// Kernel 1: fold W (28 x 256) into V (1024) : V[d*128+e] = sum over pairs.
// Tiny; launched with 1024 threads total.
// ---------------------------------------------------------------------------
__global__ __launch_bounds__(256) void fold_weights(const float* __restrict__ W,
                                                    float* __restrict__ V) {
    int t = blockIdx.x * blockDim.x + threadIdx.x;   // 0..1023
    if (t >= DTOT) return;
    int d = t >> 7;          // domain 0..7
    int e = t & 127;         // embedding index
    float s = 0.0f;
    int k = 0;
#pragma unroll
    for (int i = 0; i < ND; ++i) {
#pragma unroll
        for (int j = i + 1; j < ND; ++j) {
            if (i == d) s += W[k * 256 + e];           // Wi half
            if (j == d) s += W[k * 256 + 128 + e];     // Wj half
            ++k;
        }
    }
    V[t] = s;
}

// ---------------------------------------------------------------------------
// Kernel 2: memory-bound GEMV out[b] = x[b,:] . V  via V_WMMA_F32_16X16X4_F32.
// Each wave: 16 rows. A = 16x4 x-chunk, B = 4x16 with every column = V chunk,
// so every column of the chained accumulator holds the full dot product.
// ---------------------------------------------------------------------------
__global__ __launch_bounds__(256) void gemv_wmma_f32(const float* __restrict__ x,
                                                     const float* __restrict__ V,
                                                     float* __restrict__ out) {
    __shared__ float lx[ROWS_PER_BLK * LDS_STRIDE];  // 34816 B
    __shared__ float lv[DTOT];                       //  4096 B

    const int t       = threadIdx.x;
    const int rowBase = blockIdx.x * ROWS_PER_BLK;

    // Stage folded weights once per block (hits L2, 4 KB).
    for (int i = t; i < DTOT; i += 256) lv[i] = V[i];

    const int w    = t >> 5;        // wave id 0..7
    const int lane = t & 31;
    const int m    = lane & 15;     // row within the wave's 16-row stripe
    const int h    = lane >> 4;     // half-wave: selects K {0,1} vs {2,3}

    v8f acc = {};                   // fp32 16x16 accumulator (8 VGPRs)

    for (int kt = 0; kt < NKTILES; ++kt) {
        __syncthreads();            // previous tile's readers done
        // --- coalesced global -> LDS: 128 rows x 64 floats (b128 per lane) ---
#pragma unroll
        for (int i = 0; i < (ROWS_PER_BLK * KTILE / 4) / 256; ++i) {  // 8 iters
            int f   = t + i * 256;
            int r   = f >> 4;       // row in tile (16 float4 per row)
            int kk4 = f & 15;       // float4 index within the 64-float row seg
            v4f val = *(const v4f*)(x + (size_t)(rowBase + r) * DTOT
                                      + kt * KTILE + kk4 * 4);
            *(v4f*)(&lx[r * LDS_STRIDE + kk4 * 4]) = val;
        }
        __syncthreads();

        // --- 16 chained K=4 WMMA steps out of LDS (conflict-free layout) ---
        const float* arow  = &lx[(w * 16 + m) * LDS_STRIDE + 2 * h];
        const float* vbase = &lv[kt * KTILE + 2 * h];
#pragma unroll
        for (int c = 0; c < CHUNKS; ++c) {
            v2f a = *(const v2f*)(arow  + c * 4);   // A: x[m][k..k+1] per half
            v2f b = *(const v2f*)(vbase + c * 4);   // B: V chunk broadcast to all N
            acc = __builtin_amdgcn_wmma_f32_16x16x4_f32(
                /*neg_a=*/false, a, /*neg_b=*/false, b,
                /*c_mod=*/(short)0, acc, /*reuse_a=*/false, /*reuse_b=*/false);
        }
    }

    // D layout: VGPR j, lanes 0-15 -> M=j ; lanes 16-31 -> M=8+j. Column N=0/16.
    if (m == 0) {
        int base = rowBase + w * 16 + h * 8;
#pragma unroll
        for (int j = 0; j < 8; ++j) out[base + j] = acc[j];
    }
}

// ---------------------------------------------------------------------------
extern "C" void kernel_launch(void* const* d_in, const int* in_sizes, int n_in,
                              void* d_out, int out_size, void* d_ws, size_t ws_size,
                              hipStream_t stream) {
    const float* x = (const float*)d_in[0];   // (B, 8, 128) fp32
    const float* W = (const float*)d_in[1];   // (28, 256) fp32
    float*       o = (float*)d_out;           // (B,) fp32
    float*       V = (float*)d_ws;            // 1024 floats of scratch

    const int B = in_sizes[0] / DTOT;         // 131072

    fold_weights<<<(DTOT + 255) / 256, 256, 0, stream>>>(W, V);
    gemv_wmma_f32<<<B / ROWS_PER_BLK, 256, 0, stream>>>(x, V, o);
}